// kMeanCluster_84095459656141
// MI455X (gfx1250) — compile-verified
//
#include <hip/hip_runtime.h>
#include <stdint.h>

// ---------------------------------------------------------------------------
// Squared-euclidean-distance-to-centers for MI455X / gfx1250.
// dist[m,n] = ||f_m||^2 + ||c_n||^2 - 2 * <f_m, c_n>.
// Cross term: V_WMMA_F32_16X16X4_F32 (fp32 matrix cores, wave32).
// Staging:    GLOBAL_LOAD_ASYNC_TO_LDS_B128, double-buffered LDS (ASYNCcnt).
// ---------------------------------------------------------------------------

typedef __attribute__((ext_vector_type(2))) float v2f_t;
typedef __attribute__((ext_vector_type(8))) float v8f_t;

#define D_DIM   256     // feature dim (reduction)
#define K_TOT   512     // number of centers (N)
#define M_TOT   32768   // B*S = 16*2048 (M)
#define BM      64      // block tile M
#define BN      128     // block tile N
#define KC      64      // K-chunk staged in LDS
#define NCHUNK  (D_DIM / KC)
#define LDST    68      // LDS row stride (dwords): +4 banks/row -> conflict-free frags

static __device__ __forceinline__ void wait_asynccnt0() {
#if __has_builtin(__builtin_amdgcn_s_wait_asynccnt)
    __builtin_amdgcn_s_wait_asynccnt(0);
#else
    asm volatile("s_wait_asynccnt 0x0" ::: "memory");
#endif
}

// One b128 async copy: 16 bytes from global address base + goff (bytes) to the LDS byte address lds_byte
static __device__ __forceinline__ void async_cp_b128(unsigned lds_byte,
                                                     unsigned goff,
                                                     const float* base) {
    asm volatile("global_load_async_to_lds_b128 %0, %1, %2"
                 :: "v"(lds_byte), "v"(goff), "s"(base)
                 : "memory");
}

// ---------------------------- prologue kernels -----------------------------

__global__ __launch_bounds__(256)
void kmc_fsq_kernel(const float* __restrict__ F, float* __restrict__ fsq) {
    const int m = blockIdx.x * 256 + threadIdx.x;          // 128 blocks -> 32768
    const float4* row = (const float4*)(F + (size_t)m * D_DIM);
    float s = 0.0f;
#pragma unroll 8
    for (int j = 0; j < D_DIM / 4; ++j) {
        float4 v = row[j];
        s += v.x * v.x + v.y * v.y + v.z * v.z + v.w * v.w;
    }
    fsq[m] = s;
}

__global__ __launch_bounds__(256)
void kmc_csq_kernel(const float* __restrict__ C, float* __restrict__ csq) {
    const int n = blockIdx.x * 256 + threadIdx.x;          // 2 blocks -> 512
    const float4* row = (const float4*)(C + (size_t)n * D_DIM);
    float s = 0.0f;
#pragma unroll 8
    for (int j = 0; j < D_DIM / 4; ++j) {
        float4 v = row[j];
        s += v.x * v.x + v.y * v.y + v.z * v.z + v.w * v.w;
    }
    csq[n] = s;
}

// ------------------------------ main kernel --------------------------------
// grid = (M_TOT/BM, K_TOT/BN) = (512, 4); block = 256 threads = 8 waves.
// Wave (wm in 0..3, wn in 0..1): rows [m0+16*wm, +16), cols [n0+64*wn, +64).

__global__ __launch_bounds__(256)
void kmc_dist_wmma_kernel(const float* __restrict__ F,
                          const float* __restrict__ C,
                          const float* __restrict__ fsq,
                          const float* __restrict__ csq,
                          float* __restrict__ out) {
    __shared__ float sA[2][BM * LDST];   // 2 x (64 x 64) features chunk (padded)
    __shared__ float sB[2][BN * LDST];   // 2 x (128 x 64) centers  chunk (padded)

    const int t    = threadIdx.x;
    const int lane = t & 31;             // wave32
    const int w    = t >> 5;
    const int wm   = w & 3;
    const int wn   = w >> 2;
    const int lo   = lane & 15;          // row-within-tile lane
    const int hi   = lane >> 4;          // K-half selector
    const int m0   = blockIdx.x * BM;
    const int n0   = blockIdx.y * BN;

    // Per-thread staging coordinates (same for every chunk, only kc varies):
    // A: 4 float4s/thread over 64x64; B: 8 float4s/thread over 128x64.
    const int r16 = t >> 4;              // 0..15
    const int c4  = (t & 15) << 2;       // 0,4,...,60

    v8f_t acc[4];
    acc[0] = {}; acc[1] = {}; acc[2] = {}; acc[3] = {};

    // ---- issue async staging of chunk `kc` into buffer `b` -----------------
    auto stage = [&](int b, int kc) {
#pragma unroll
        for (int i = 0; i < 4; ++i) {
            const int r = r16 + 16 * i;                       // 0..63
            const unsigned lds  = (unsigned)(uintptr_t)(&sA[b][r * LDST + c4]);
            const unsigned goff = (unsigned)(((m0 + r) * D_DIM + kc + c4) * 4);
            async_cp_b128(lds, goff, F);
        }
#pragma unroll
        for (int i = 0; i < 8; ++i) {
            const int r = r16 + 16 * i;                       // 0..127
            const unsigned lds  = (unsigned)(uintptr_t)(&sB[b][r * LDST + c4]);
            const unsigned goff = (unsigned)(((n0 + r) * D_DIM + kc + c4) * 4);
            async_cp_b128(lds, goff, C);
        }
    };

    // ---- prologue: prefetch chunk 0 ---------------------------------------
    stage(0, 0);
    wait_asynccnt0();
    __syncthreads();

    // Per-wave fragment base offsets inside a buffer (dword units)
    const int aFrag = (wm * 16 + lo) * LDST + 2 * hi;
    const int bFrag = (wn * 64 + lo) * LDST + 2 * hi;

    // ---- main pipeline: prefetch next chunk while computing current -------
#pragma unroll
    for (int ck = 0; ck < NCHUNK; ++ck) {
        const int cur = ck & 1;
        if (ck + 1 < NCHUNK) stage(cur ^ 1, (ck + 1) * KC);

        const float* a_base = &sA[cur][aFrag];
        const float* b_base = &sB[cur][bFrag];
#pragma unroll
        for (int ks = 0; ks < KC; ks += 4) {
            // A fragment: lane -> (row = lane&15, K = ks + 2*hi + {0,1})
            v2f_t a = *(const v2f_t*)(a_base + ks);
#pragma unroll
            for (int nt = 0; nt < 4; ++nt) {
                // B fragment: lane -> (N = lane&15, K = ks + 2*hi + {0,1});
                // Ck row-major over D is exactly K-consecutive for fixed N.
                v2f_t b = *(const v2f_t*)(b_base + nt * 16 * LDST + ks);
                acc[nt] = __builtin_amdgcn_wmma_f32_16x16x4_f32(
                    /*neg_a=*/false, a, /*neg_b=*/false, b,
                    /*c_mod=*/(short)0, acc[nt],
                    /*reuse_a=*/false, /*reuse_b=*/false);
            }
        }
        wait_asynccnt0();   // next chunk's async copies landed in LDS
        __syncthreads();    // all waves done reading `cur`, writes visible
    }

    // --- epilogue: out[m,n] = fsq[m] + csq[n] - 2*acc ----------------------
    const int mrow  = m0 + wm * 16 + hi * 8;       // + i per C VGPR
    const int ncol0 = n0 + wn * 64 + lo;

    float cn[4];
#pragma unroll
    for (int nt = 0; nt < 4; ++nt) cn[nt] = csq[ncol0 + nt * 16];

#pragma unroll
    for (int i = 0; i < 8; ++i) {
        const int m = mrow + i;
        const float fm = fsq[m];
#pragma unroll
        for (int nt = 0; nt < 4; ++nt) {
            const int n = ncol0 + nt * 16;
            float val = fm + cn[nt] - 2.0f * acc[nt][i];
            // write-once 67 MB stream: non-temporal, keep L2 for F/C
            __builtin_nontemporal_store(val, &out[(size_t)m * K_TOT + n]);
        }
    }
}

// ------------------------------- launcher ----------------------------------

extern "C" void kernel_launch(void* const* d_in, const int* in_sizes, int n_in,
                              void* d_out, int out_size, void* d_ws, size_t ws_size,
                              hipStream_t stream) {
    const float* features = (const float*)d_in[0];   // (16,2048,256) f32
    const float* Ck       = (const float*)d_in[1];   // (1,512,256)   f32
    float* out = (float*)d_out;                      // (16,2048,512) f32

    float* fsq = (float*)d_ws;                       // 32768 f32
    float* csq = fsq + M_TOT;                        // 512 f32

    kmc_fsq_kernel<<<M_TOT / 256, 256, 0, stream>>>(features, fsq);
    kmc_csq_kernel<<<K_TOT / 256, 256, 0, stream>>>(Ck, csq);

    dim3 grid(M_TOT / BM, K_TOT / BN, 1);            // (512, 4)
    kmc_dist_wmma_kernel<<<grid, 256, 0, stream>>>(features, Ck, fsq, csq, out);
}